// RasterizeSil_15625091023489
// MI455X (gfx1250) — compile-verified
//
#include <hip/hip_runtime.h>

#define IMG 256
#define NEARP 0.1f
#define FARP 100.0f
#define EPSX 0.001f

typedef float v2f __attribute__((ext_vector_type(2)));
typedef float v8f __attribute__((ext_vector_type(8)));

// ---------------------------------------------------------------------------
// Precompute per-face affine coefficients, sign-normalized and pre-divided by
// denom, laid out exactly in V_WMMA_F32_16X16X4_F32 A-operand order:
//   group of 4 faces -> 64 floats; flat index = lane*2 + (k&1)
//   lane m    (m=4*faceInGroup+q) holds K0 (px coef) and K1 (py coef)
//   lane m+16 holds K2 (const coef) and K3 (=0)
// quantities q: 0..2 = edge functions, 3 = depth (zp) -- zp comes straight
// out of the WMMA, no division needed in the hot loop.
// ---------------------------------------------------------------------------
__global__ void RasterSil_precomp(const float* __restrict__ faces,
                                  float* __restrict__ coefs,
                                  int totalFaces) {
    int f = blockIdx.x * blockDim.x + threadIdx.x;
    if (f >= totalFaces) return;
    const float* p = faces + (size_t)f * 9;
    float x0 = p[0], y0 = p[1], z0 = p[2];
    float x1 = p[3], y1 = p[4], z1 = p[5];
    float x2 = p[6], y2 = p[7], z2 = p[8];

    // w_e(px,py) = A_e*px + B_e*py + C_e
    float A0 = y1 - y2, B0 = x2 - x1, C0 = x1 * y2 - x2 * y1;
    float A1 = y2 - y0, B1 = x0 - x2, C1 = x2 * y0 - x0 * y2;
    float A2 = y0 - y1, B2 = x1 - x0, C2 = x0 * y1 - x1 * y0;

    float area = C0 + C1 + C2;                 // w0+w1+w2 is constant per face
    float s = (area >= 0.0f) ? 1.0f : -1.0f;   // flip so "inside" == all >= 0
    float scale = s / (fabsf(area) + EPSX);    // fold denom into coefficients

    float qa[4] = {A0, A1, A2, z0 * A0 + z1 * A1 + z2 * A2};
    float qb[4] = {B0, B1, B2, z0 * B0 + z1 * B1 + z2 * B2};
    float qc[4] = {C0, C1, C2, z0 * C0 + z1 * C1 + z2 * C2};

    int g = f >> 2;      // global 4-face group (batch folded in)
    int j = f & 3;
    float* base = coefs + (size_t)g * 64;
#pragma unroll
    for (int q = 0; q < 4; ++q) {
        int m = 4 * j + q;
        base[m * 2 + 0]        = qa[q] * scale;  // K0
        base[m * 2 + 1]        = qb[q] * scale;  // K1
        base[(m + 16) * 2 + 0] = qc[q] * scale;  // K2
        base[(m + 16) * 2 + 1] = 0.0f;           // K3 (multiplies pixel 0-lane)
    }
}

// ---------------------------------------------------------------------------
// Main rasterizer: each wave owns 16 pixels of one image row; loops over all
// 4-face groups. One f32 16x16x4 WMMA evaluates {w0,w1,w2,zp} for 4 faces x
// 16 pixels. Lanes 0-15 test faces 0-1, lanes 16-31 test faces 2-3 of each
// group (all lane-local thanks to the C/D VGPR layout); a single shfl_xor(16)
// merges the two coverage halves at the end.
// ---------------------------------------------------------------------------
__global__ __launch_bounds__(256) void RasterSil_main(
        const float* __restrict__ coefs, float* __restrict__ out,
        int groupsPerBatch) {
    const int lane = threadIdx.x & 31;
    const int wave = blockIdx.x * (blockDim.x >> 5) + (threadIdx.x >> 5);
    const int b    = wave >> 12;      // 4096 waves per batch (256 rows x 16)
    const int rem  = wave & 4095;
    const int y    = rem >> 4;
    const int xg   = rem & 15;
    const int n    = lane & 15;       // pixel column within the 16-pixel tile

    const float pxv = (2.0f * (float)(xg * 16 + n) + 1.0f - (float)IMG) / (float)IMG;
    const float pyv = -(2.0f * (float)y + 1.0f - (float)IMG) / (float)IMG;

    // B operand (4x16): column n = (px, py, 1, 0)
    v2f bmat;
    if (lane < 16) { bmat.x = pxv;  bmat.y = pyv;  }   // rows K0,K1
    else           { bmat.x = 1.0f; bmat.y = 0.0f; }   // rows K2,K3

    const float* cb = coefs + (size_t)(b * groupsPerBatch) * 64 + (size_t)lane * 2;

    float cov = 0.0f;
#pragma unroll 2
    for (int g = 0; g < groupsPerBatch; ++g) {
        v2f amat = *(const v2f*)(cb + (size_t)g * 64);   // global_load_b64
        v8f acc = {};
        acc = __builtin_amdgcn_wmma_f32_16x16x4_f32(
            /*neg_a=*/false, amat, /*neg_b=*/false, bmat,
            /*c_mod=*/(short)0, acc, /*reuse_a=*/false, /*reuse_b=*/false);
#pragma unroll
        for (int jj = 0; jj < 2; ++jj) {
            float w0 = acc[4 * jj + 0];
            float w1 = acc[4 * jj + 1];
            float w2 = acc[4 * jj + 2];
            float zp = acc[4 * jj + 3];
            bool vis = (w0 >= 0.0f) & (w1 >= 0.0f) & (w2 >= 0.0f) &
                       (zp > NEARP) & (zp < FARP);
            cov = vis ? 1.0f : cov;
        }
    }

    // merge the two half-wave face partitions (same 16 pixels)
    float other = __shfl_xor(cov, 16, 32);
    cov = fmaxf(cov, other);

    if (lane < 16) {
        out[(size_t)b * (IMG * IMG) + (size_t)y * IMG + xg * 16 + n] = cov;
    }
}

extern "C" void kernel_launch(void* const* d_in, const int* in_sizes, int n_in,
                              void* d_out, int out_size, void* d_ws, size_t ws_size,
                              hipStream_t stream) {
    const float* faces = (const float*)d_in[0];
    float* out   = (float*)d_out;
    float* coefs = (float*)d_ws;                 // 16 floats/face = 1 MB total

    const int totalFaces     = in_sizes[0] / 9;                // B*F
    const int B              = out_size / (IMG * IMG);
    const int Fperm          = totalFaces / B;
    const int groupsPerBatch = Fperm / 4;

    RasterSil_precomp<<<(totalFaces + 255) / 256, 256, 0, stream>>>(
        faces, coefs, totalFaces);

    const int wavesTotal = B * (IMG * IMG / 16);               // 16384
    RasterSil_main<<<wavesTotal / 8, 256, 0, stream>>>(
        coefs, out, groupsPerBatch);
}